// MambaBlock_24240795418673
// MI455X (gfx1250) — compile-verified
//
#include <hip/hip_runtime.h>
#include <math.h>

// ---------------- problem constants ----------------
constexpr int DIM     = 1024;
constexpr int D_INNER = 2048;
constexpr int D_STATE = 16;
constexpr int D_CONV  = 4;
constexpr int DT_RANK = 64;
constexpr int B_SZ    = 2;
constexpr int T_LEN   = 2048;
constexpr int M_TOK   = B_SZ * T_LEN;          // 4096 tokens
constexpr int PROJ_W  = DT_RANK + 2 * D_STATE; // 96

typedef __attribute__((ext_vector_type(2)))  float v2f;
typedef __attribute__((ext_vector_type(8)))  float v8f;

__device__ __forceinline__ float softplus_f(float x) {
  return (x > 20.f) ? x : log1pf(__expf(x));
}
__device__ __forceinline__ float silu_f(float x) {
  return x / (1.f + __expf(-x));
}

// ---------------------------------------------------------------------------
// Blocked f32 WMMA GEMM:  C[M,N] = A[M,K] @ B[K,N]  (row-major).
//   block  : 256 threads / 8 waves -> 128x128 macro-tile
//   wave   : 64x32 region = 4(M) x 2(N) V_WMMA_F32_16X16X4_F32 accumulators
//   K path : double-buffered LDS, KB=16 per stage
//     As[128][20] row-major (pad->conflict-free; 80B rows keep float4 stores
//                 16B-aligned and make each A fragment one ds_load_b64)
//     Bs[16][132] k-major   (coalesced float4 staging, conflict-free b32 reads)
// epilogue: 0 = none, 1 = softplus(acc + bias[col])
// ---------------------------------------------------------------------------
constexpr int KB = 16;

__global__ __launch_bounds__(256) void wmma_gemm_blocked(
    const float* __restrict__ A, int lda,
    const float* __restrict__ Bm, int ldb,
    float* __restrict__ C, int ldc,
    int M, int N, int K,
    const float* __restrict__ bias, int epilogue) {
  __shared__ float As[2][128][20];
  __shared__ float Bs[2][KB][132];

  const int nbn = N >> 7;                     // N / 128
  const int bm  = blockIdx.x / nbn;
  const int bn  = blockIdx.x - bm * nbn;
  const int tid = threadIdx.x;
  const int wid = tid >> 5;
  const int wm  = wid >> 2;                   // 0..1 -> 64-row half
  const int wn  = wid & 3;                    // 0..3 -> 32-col strip
  const int lane = tid & 31;
  const int hp   = lane >> 4;
  const int l15  = lane & 15;

  // staging index decomposition (2 float4 of A + 2 float4 of B per thread)
  const int amA0 = tid >> 2,          akq0 = tid & 3;          // A elem 0
  const int amA1 = (tid + 256) >> 2,  akq1 = (tid + 256) & 3;  // A elem 1
  const int bkk0 = tid >> 5,          bnq0 = tid & 31;         // B elem 0
  const int bkk1 = (tid + 256) >> 5,  bnq1 = (tid + 256) & 31; // B elem 1

  const float* Abase = A + (size_t)(bm * 128) * lda;
  const float* Bbase = Bm + bn * 128;

  v8f acc[4][2];
  #pragma unroll
  for (int sm = 0; sm < 4; ++sm)
    #pragma unroll
    for (int sn = 0; sn < 2; ++sn)
      acc[sm][sn] = (v8f){};

  const int nkb = K / KB;

  // ---- prologue: stage kb = 0 into buffer 0 ----
  float4 a0 = *(const float4*)&Abase[(size_t)amA0 * lda + akq0 * 4];
  float4 a1 = *(const float4*)&Abase[(size_t)amA1 * lda + akq1 * 4];
  float4 b0 = *(const float4*)&Bbase[(size_t)bkk0 * ldb + bnq0 * 4];
  float4 b1 = *(const float4*)&Bbase[(size_t)bkk1 * ldb + bnq1 * 4];
  *(float4*)&As[0][amA0][akq0 * 4] = a0;
  *(float4*)&As[0][amA1][akq1 * 4] = a1;
  *(float4*)&Bs[0][bkk0][bnq0 * 4] = b0;
  *(float4*)&Bs[0][bkk1][bnq1 * 4] = b1;

  int buf = 0;
  for (int kbi = 0; kbi < nkb; ++kbi) {
    __syncthreads();
    const bool have_next = (kbi + 1) < nkb;
    if (have_next) {                          // issue next-stage global loads
      const int kb = (kbi + 1) * KB;
      a0 = *(const float4*)&Abase[(size_t)amA0 * lda + kb + akq0 * 4];
      a1 = *(const float4*)&Abase[(size_t)amA1 * lda + kb + akq1 * 4];
      b0 = *(const float4*)&Bbase[(size_t)(kb + bkk0) * ldb + bnq0 * 4];
      b1 = *(const float4*)&Bbase[(size_t)(kb + bkk1) * ldb + bnq1 * 4];
      if (kbi + 2 < nkb) {                    // stream hint two stages ahead
        const int kp = (kbi + 2) * KB;
        __builtin_prefetch(&Abase[(size_t)amA0 * lda + kp], 0, 0);
        __builtin_prefetch(&Bbase[(size_t)(kp + bkk0) * ldb + bnq0 * 4], 0, 0);
      }
    }
    // ---- compute on current buffer: 4 k-steps x 8 wmma ----
    #pragma unroll
    for (int kk = 0; kk < KB; kk += 4) {
      v2f af[4], bf[2];
      #pragma unroll
      for (int sm = 0; sm < 4; ++sm) {
        const float* p = &As[buf][wm * 64 + sm * 16 + l15][kk + 2 * hp];
        af[sm].x = p[0];
        af[sm].y = p[1];
      }
      #pragma unroll
      for (int sn = 0; sn < 2; ++sn) {
        bf[sn].x = Bs[buf][kk + 2 * hp][wn * 32 + sn * 16 + l15];
        bf[sn].y = Bs[buf][kk + 2 * hp + 1][wn * 32 + sn * 16 + l15];
      }
      #pragma unroll
      for (int sm = 0; sm < 4; ++sm)
        #pragma unroll
        for (int sn = 0; sn < 2; ++sn)
          acc[sm][sn] = __builtin_amdgcn_wmma_f32_16x16x4_f32(
              false, af[sm], false, bf[sn], (short)0, acc[sm][sn], false, false);
    }
    if (have_next) {                          // store next stage to other buffer
      const int nb = buf ^ 1;
      *(float4*)&As[nb][amA0][akq0 * 4] = a0;
      *(float4*)&As[nb][amA1][akq1 * 4] = a1;
      *(float4*)&Bs[nb][bkk0][bnq0 * 4] = b0;
      *(float4*)&Bs[nb][bkk1][bnq1 * 4] = b1;
    }
    buf ^= 1;
  }

  // ---- epilogue ----
  #pragma unroll
  for (int sn = 0; sn < 2; ++sn) {
    const int col = bn * 128 + wn * 32 + sn * 16 + l15;
    const float bv = (epilogue == 1) ? bias[col] : 0.f;
    #pragma unroll
    for (int sm = 0; sm < 4; ++sm) {
      #pragma unroll
      for (int r = 0; r < 8; ++r) {
        const int row = bm * 128 + wm * 64 + sm * 16 + r + 8 * hp;
        float v = acc[sm][sn][r];
        if (epilogue == 1) v = softplus_f(v + bv);
        C[(size_t)row * ldc + col] = v;
      }
    }
  }
}

// ---------------------------------------------------------------------------
// Simple WMMA GEMM (one 16x16 tile per wave) for skinny N (proj: N=96).
// ---------------------------------------------------------------------------
__global__ __launch_bounds__(256) void wmma_gemm_simple(
    const float* __restrict__ A, int lda,
    const float* __restrict__ Bm, int ldb,
    float* __restrict__ C, int ldc,
    int M, int N, int K) {
  const int ntn   = N >> 4;
  const int tiles = (M >> 4) * ntn;
  const int tile  = blockIdx.x * 8 + (threadIdx.x >> 5);
  if (tile >= tiles) return;                 // uniform per wave -> EXEC all-1s
  const int tm   = tile / ntn;
  const int tn   = tile - tm * ntn;
  const int lane = threadIdx.x & 31;
  const int hp   = lane >> 4;
  const int l15  = lane & 15;
  const float* Arow = A + (size_t)(tm * 16 + l15) * lda;
  const float* Bcol = Bm + tn * 16 + l15;
  v8f c = {};
  #pragma unroll 4
  for (int k = 0; k < K; k += 4) {
    v2f a, b;
    a.x = Arow[k + 2 * hp];
    a.y = Arow[k + 2 * hp + 1];
    b.x = Bcol[(size_t)(k + 2 * hp) * ldb];
    b.y = Bcol[(size_t)(k + 2 * hp + 1) * ldb];
    c = __builtin_amdgcn_wmma_f32_16x16x4_f32(false, a, false, b,
                                              (short)0, c, false, false);
  }
  const int col = tn * 16 + l15;
  #pragma unroll
  for (int r = 0; r < 8; ++r)
    C[(size_t)(tm * 16 + r + 8 * hp) * ldc + col] = c[r];
}

// ---------------------------------------------------------------------------
// Depthwise causal conv (k=4) + bias + SiLU.
// ---------------------------------------------------------------------------
__global__ __launch_bounds__(256) void conv_silu_kernel(
    const float* __restrict__ xz,          // x_in slice, ld = 2*D_INNER
    const float* __restrict__ conv_state,  // [B, D_INNER, 3]
    const float* __restrict__ conv_w,      // [D_INNER, 4]
    const float* __restrict__ conv_b,      // [D_INNER]
    float* __restrict__ x_conv) {          // [M_TOK, D_INNER]
  const size_t idx = (size_t)blockIdx.x * 256 + threadIdx.x;
  const int d = (int)(idx & (D_INNER - 1));
  const int t = (int)((idx >> 11) & (T_LEN - 1));
  const int b = (int)(idx >> 22);
  float acc = conv_b[d];
  #pragma unroll
  for (int j = 0; j < D_CONV; ++j) {
    const int tin = t + j - (D_CONV - 1);
    const float v = (tin < 0)
        ? conv_state[((size_t)(b * D_INNER + d)) * (D_CONV - 1) + (tin + D_CONV - 1)]
        : xz[((size_t)(b * T_LEN + tin)) * (2 * D_INNER) + d];
    acc += conv_w[d * D_CONV + j] * v;
  }
  x_conv[idx] = silu_f(acc);
}

// new_conv_state = last 3 timesteps of x_in, transposed to [B, D_INNER, 3]
__global__ __launch_bounds__(256) void conv_state_out_kernel(
    const float* __restrict__ xz, float* __restrict__ ncs) {
  const int idx = blockIdx.x * 256 + threadIdx.x;
  if (idx >= B_SZ * D_INNER * (D_CONV - 1)) return;
  const int j = idx % 3;
  const int d = (idx / 3) % D_INNER;
  const int b = idx / (3 * D_INNER);
  ncs[idx] = xz[((size_t)(b * T_LEN + T_LEN - 3 + j)) * (2 * D_INNER) + d];
}

// ---------------------------------------------------------------------------
// Selective-scan: one thread owns one (b,d) channel, 16 states in VGPRs.
// B_t / C_t staged in LDS in 32-timestep chunks; writes y gated by silu(z).
// ---------------------------------------------------------------------------
__global__ __launch_bounds__(256) void mamba_scan_kernel(
    const float* __restrict__ dt,      // [M_TOK, D_INNER]
    const float* __restrict__ xconv,   // [M_TOK, D_INNER]
    const float* __restrict__ proj,    // [M_TOK, 96]
    const float* __restrict__ zbuf,    // z slice of xz, ld = 2*D_INNER
    const float* __restrict__ A_log,   // [D_INNER, 16]
    const float* __restrict__ Dskip,   // [D_INNER]
    const float* __restrict__ h0,      // [B, D_INNER, 16]
    float* __restrict__ y,             // [M_TOK, D_INNER]
    float* __restrict__ h_final) {     // [B, D_INNER, 16]
  const int b = blockIdx.x >> 3;
  const int d = ((blockIdx.x & 7) << 8) + threadIdx.x;

  float h[D_STATE], Aneg[D_STATE];
  #pragma unroll
  for (int n = 0; n < D_STATE; ++n) {
    h[n]    = h0[((size_t)(b * D_INNER + d)) * D_STATE + n];
    Aneg[n] = -__expf(A_log[d * D_STATE + n]);
  }
  const float Dd = Dskip[d];

  __shared__ float sB[32][D_STATE];
  __shared__ float sC[32][D_STATE];

  for (int tc = 0; tc < T_LEN; tc += 32) {
    for (int i = threadIdx.x; i < 32 * 32; i += 256) {
      const int tt = i >> 5, comp = i & 31;
      const size_t prow = ((size_t)(b * T_LEN + tc + tt)) * PROJ_W;
      if (comp < D_STATE) sB[tt][comp]           = proj[prow + DT_RANK + comp];
      else                sC[tt][comp - D_STATE] = proj[prow + DT_RANK + D_STATE + (comp - D_STATE)];
    }
    __syncthreads();
    for (int tt = 0; tt < 32; ++tt) {
      const size_t tok  = (size_t)(b * T_LEN + tc + tt);
      const size_t base = tok * D_INNER + d;
      const float dtv = dt[base];
      const float xv  = xconv[base];
      const float zv  = zbuf[tok * (2 * D_INNER) + d];
      float acc = 0.f;
      #pragma unroll
      for (int n = 0; n < D_STATE; ++n) {
        const float dA = __expf(dtv * Aneg[n]);
        h[n] = dA * h[n] + (dtv * sB[tt][n]) * xv;
        acc += h[n] * sC[tt][n];
      }
      y[base] = (acc + Dd * xv) * silu_f(zv);
    }
    __syncthreads();
  }
  #pragma unroll
  for (int n = 0; n < D_STATE; ++n)
    h_final[((size_t)(b * D_INNER + d)) * D_STATE + n] = h[n];
}

// ---------------------------------------------------------------------------
extern "C" void kernel_launch(void* const* d_in, const int* in_sizes, int n_in,
                              void* d_out, int out_size, void* d_ws, size_t ws_size,
                              hipStream_t stream) {
  (void)in_sizes; (void)n_in; (void)out_size; (void)ws_size;
  const float* x          = (const float*)d_in[0];
  const float* ssm0       = (const float*)d_in[1];
  const float* conv_state = (const float*)d_in[2];
  const float* w_in       = (const float*)d_in[3];
  const float* conv_w     = (const float*)d_in[4];
  const float* conv_b     = (const float*)d_in[5];
  const float* w_x        = (const float*)d_in[6];
  const float* w_dt       = (const float*)d_in[7];
  const float* b_dt       = (const float*)d_in[8];
  const float* A_log      = (const float*)d_in[9];
  const float* D_skip     = (const float*)d_in[10];
  const float* w_out      = (const float*)d_in[11];

  float* out     = (float*)d_out;                              // [M_TOK, DIM]
  float* h_final = out + (size_t)M_TOK * DIM;                  // [B, D_INNER, 16]
  float* ncs     = h_final + (size_t)B_SZ * D_INNER * D_STATE; // [B, D_INNER, 3]

  float* ws    = (float*)d_ws;
  float* xz    = ws;                                   // [M_TOK, 2*D_INNER]
  float* xconv = xz    + (size_t)M_TOK * 2 * D_INNER;  // [M_TOK, D_INNER]
  float* proj  = xconv + (size_t)M_TOK * D_INNER;      // [M_TOK, 96]
  float* dtb   = proj  + (size_t)M_TOK * PROJ_W;       // [M_TOK, D_INNER]
  float* yb    = dtb   + (size_t)M_TOK * D_INNER;      // [M_TOK, D_INNER]
  float* zbuf  = xz + D_INNER;                         // z column slice

  // 1) xz = x @ w_in       (4096 x 4096 x 1024)
  wmma_gemm_blocked<<<(M_TOK / 128) * ((2 * D_INNER) / 128), 256, 0, stream>>>(
      x, DIM, w_in, 2 * D_INNER, xz, 2 * D_INNER,
      M_TOK, 2 * D_INNER, DIM, nullptr, 0);
  // 2) depthwise conv + SiLU ; new conv state
  conv_silu_kernel<<<(M_TOK * D_INNER) / 256, 256, 0, stream>>>(
      xz, conv_state, conv_w, conv_b, xconv);
  conv_state_out_kernel<<<(B_SZ * D_INNER * (D_CONV - 1) + 255) / 256, 256, 0, stream>>>(
      xz, ncs);
  // 3) proj = x_conv @ w_x  (4096 x 96 x 2048)  -- skinny N, simple kernel
  {
    const int tiles = (M_TOK / 16) * (PROJ_W / 16); // 1536
    wmma_gemm_simple<<<tiles / 8, 256, 0, stream>>>(
        xconv, D_INNER, w_x, PROJ_W, proj, PROJ_W, M_TOK, PROJ_W, D_INNER);
  }
  // 4) dt = softplus(dt_raw @ w_dt + b_dt)  (4096 x 2048 x 64)
  wmma_gemm_blocked<<<(M_TOK / 128) * (D_INNER / 128), 256, 0, stream>>>(
      proj, PROJ_W, w_dt, D_INNER, dtb, D_INNER,
      M_TOK, D_INNER, DT_RANK, b_dt, 1);
  // 5) selective scan (+ SiLU(z) gating) -> y, h_final
  mamba_scan_kernel<<<B_SZ * (D_INNER / 256), 256, 0, stream>>>(
      dtb, xconv, proj, zbuf, A_log, D_skip, ssm0, yb, h_final);
  // 6) out = y @ w_out      (4096 x 1024 x 2048)
  wmma_gemm_blocked<<<(M_TOK / 128) * (DIM / 128), 256, 0, stream>>>(
      yb, D_INNER, w_out, DIM, out, DIM, M_TOK, DIM, D_INNER, nullptr, 0);
}